// GraphSAGE_6176162971851
// MI455X (gfx1250) — compile-verified
//
#include <hip/hip_runtime.h>
#include <hip/hip_bf16.h>

#define N_NODES 50000
#define N_EDGES 800000
#define D_FEAT  128

typedef __attribute__((ext_vector_type(2))) float v2f;
typedef __attribute__((ext_vector_type(8))) float v8f;

// ---------------- zero agg + deg (float4 stores) ----------------
__global__ void sage_zero(float4* __restrict__ p, int n4) {
    int i = blockIdx.x * blockDim.x + threadIdx.x;
    if (i < n4) p[i] = make_float4(0.f, 0.f, 0.f, 0.f);
}

// ---------------- scatter-add: agg[dst] += in[src], deg[dst] += 1 ----------------
// 256 threads/block = 2 edges/block (128 threads per edge -> coalesced row copy).
// Agent-scope relaxed fetch_add -> native global_atomic_add_f32 (no CAS loop).
__global__ void sage_scatter(const float* __restrict__ in,
                             const int* __restrict__ src,
                             const int* __restrict__ dst,
                             float* __restrict__ agg,
                             float* __restrict__ deg) {
    int e = blockIdx.x * 2 + (threadIdx.x >> 7);
    int d = threadIdx.x & 127;
    if (e >= N_EDGES) return;
    int s = src[e];
    int t = dst[e];
    __hip_atomic_fetch_add(&agg[t * D_FEAT + d], in[s * D_FEAT + d],
                           __ATOMIC_RELAXED, __HIP_MEMORY_SCOPE_AGENT);
    if (d == 0)
        __hip_atomic_fetch_add(&deg[t], 1.0f,
                               __ATOMIC_RELAXED, __HIP_MEMORY_SCOPE_AGENT);
}

// ---------------- mean: agg[i] /= max(deg[row], 1) (exact IEEE division) ----------------
__global__ void sage_mean(float* __restrict__ agg, const float* __restrict__ deg) {
    int i = blockIdx.x * blockDim.x + threadIdx.x;
    if (i < N_NODES * D_FEAT) {
        float dv = fmaxf(deg[i >> 7], 1.0f);   // i / 128
        agg[i] = agg[i] / dv;
    }
}

// ---------------- fused dual-GEMM: out = act(mean@Wl + x@Wr + b) ----------------
// block = 256 threads = 8 wave32s. Block owns 16 rows; wave w owns cols [16w,16w+16).
// grid.x = N_NODES/16 = 3125 (exact). Full-precision V_WMMA_F32_16X16X4_F32.
__global__ void sage_gemm_wmma(const float* __restrict__ Amean,
                               const float* __restrict__ Ax,
                               const float* __restrict__ Wl,
                               const float* __restrict__ Wr,
                               const float* __restrict__ bias,
                               float* __restrict__ out,
                               int do_relu) {
    const int wave  = threadIdx.x >> 5;       // 0..7 -> column tile
    const int lane  = threadIdx.x & 31;
    const int lhalf = lane >> 4;              // 0: K=k,k+1 ; 1: K=k+2,k+3
    const int lmod  = lane & 15;
    const int rbase = blockIdx.x * 16;
    const int cbase = wave * 16;
    const int arow  = rbase + lmod;           // A-matrix row this lane feeds
    const int col   = cbase + lmod;           // B/C/D column this lane owns

    v8f acc = {};
    const float* amp = Amean + arow * D_FEAT + (lhalf ? 2 : 0);
    const float* axp = Ax    + arow * D_FEAT + (lhalf ? 2 : 0);

    #pragma unroll 4
    for (int k = 0; k < D_FEAT; k += 4) {
        const int ka = k + (lhalf ? 2 : 0);
        // A fragment (16x4 f32, ISA layout): lanes 0-15 -> K=k,k+1 for M=lane;
        // lanes 16-31 -> K=k+2,k+3 for M=lane-16  (8B contiguous load)
        v2f am = *(const v2f*)(amp + k);
        v2f ax = *(const v2f*)(axp + k);
        // B fragment (4x16 f32): lanes 0-15 hold rows K=k,k+1; lanes 16-31 rows K=k+2,k+3
        v2f bl, br;
        bl.x = Wl[(ka + 0) * D_FEAT + col];
        bl.y = Wl[(ka + 1) * D_FEAT + col];
        br.x = Wr[(ka + 0) * D_FEAT + col];
        br.y = Wr[(ka + 1) * D_FEAT + col];
        acc = __builtin_amdgcn_wmma_f32_16x16x4_f32(false, am, false, bl,
                                                    (short)0, acc, false, false);
        acc = __builtin_amdgcn_wmma_f32_16x16x4_f32(false, ax, false, br,
                                                    (short)0, acc, false, false);
    }

    // C/D layout: VGPR v holds M = v + 8*lhalf, N = col
    float bv = bias[col];
    #pragma unroll
    for (int v = 0; v < 8; ++v) {
        float r = acc[v] + bv;
        if (do_relu) r = fmaxf(r, 0.0f);
        int m = v + (lhalf ? 8 : 0);
        out[(rbase + m) * D_FEAT + col] = r;
    }
}

extern "C" void kernel_launch(void* const* d_in, const int* in_sizes, int n_in,
                              void* d_out, int out_size, void* d_ws, size_t ws_size,
                              hipStream_t stream) {
    const float* x    = (const float*)d_in[0];
    const int*   eidx = (const int*)d_in[1];   // [2, E]: row 0 = src, row 1 = dst
    const float* W1l  = (const float*)d_in[2];
    const float* b1   = (const float*)d_in[3];
    const float* W1r  = (const float*)d_in[4];
    const float* W2l  = (const float*)d_in[5];
    const float* b2   = (const float*)d_in[6];
    const float* W2r  = (const float*)d_in[7];
    float* out = (float*)d_out;

    const int* src = eidx;
    const int* dst = eidx + N_EDGES;

    float* agg = (float*)d_ws;                    // N*D
    float* deg = agg + (size_t)N_NODES * D_FEAT;  // N
    float* h   = deg + N_NODES;                   // N*D

    const int nzero4  = (N_NODES * D_FEAT + N_NODES) / 4;  // 6,450,000 / 4 exact
    const int zeroBl  = (nzero4 + 255) / 256;
    const int scatBl  = (N_EDGES + 1) / 2;
    const int meanBl  = (N_NODES * D_FEAT + 255) / 256;
    const int gemmBl  = N_NODES / 16;             // 3125, exact

    // ---- layer 1 ----
    sage_zero<<<zeroBl, 256, 0, stream>>>((float4*)agg, nzero4);  // agg+deg contiguous
    sage_scatter<<<scatBl, 256, 0, stream>>>(x, src, dst, agg, deg);
    sage_mean<<<meanBl, 256, 0, stream>>>(agg, deg);
    sage_gemm_wmma<<<gemmBl, 256, 0, stream>>>(agg, x, W1l, W1r, b1, h, 1);

    // ---- layer 2 ----
    sage_zero<<<zeroBl, 256, 0, stream>>>((float4*)agg, nzero4);
    sage_scatter<<<scatBl, 256, 0, stream>>>(h, src, dst, agg, deg);
    sage_mean<<<meanBl, 256, 0, stream>>>(agg, deg);
    sage_gemm_wmma<<<gemmBl, 256, 0, stream>>>(agg, h, W2l, W2r, b2, out, 0);
}